// FFTConv1d_thresh_14147622273325
// MI455X (gfx1250) — compile-verified
//
#include <hip/hip_runtime.h>
#include <math.h>

// FFT-domain conv with spectral threshold, f32 end-to-end.
// Pipeline: [A] FFT(x)+mask -> Xr/Xi[k][t*16+c]   (t padded to 16 rows, rows 8..15 = 0)
//           [B] FFT(rev(pad(W))) -> Wfr/Wfi[k][o*16+c]  (c innermost -> b64 operand loads)
//           [C] per-k complex GEMM via V_WMMA_F32_16X16X4_F32 -> Yr/Yi[t*32+o][k]
//           [D] IFFT + roll(1) + crop 62x62 + bias -> out

#define FFT_N   4096
#define FFT_LG2 12
#define NT      256
#define T_DIM   8
#define T_PAD   16          // A-matrix M dimension (rows 8..15 zero)
#define C_DIM   16
#define O_DIM   32
#define OUT_HW  62
#define K_PER_WAVE 4

typedef __attribute__((ext_vector_type(2))) float v2f;
typedef __attribute__((ext_vector_type(8))) float v8f;

__device__ __forceinline__ v2f ld2(const float* p) { return *reinterpret_cast<const v2f*>(p); }

// Radix-2 Stockham autosort FFT over LDS ping-pong buffers.
// sign = -1 forward, +1 inverse (unscaled). 12 stages (even) -> result lands back in (r0,i0).
__device__ __forceinline__ void block_fft(float* r0, float* i0, float* r1, float* i1, float sign)
{
    __syncthreads();
    float* sR = r0; float* sI = i0; float* dR = r1; float* dI = i1;
    int n = FFT_N;
    for (int stage = 0; stage < FFT_LG2; ++stage) {
        const int   s   = 1 << stage;
        const float fac = sign * 6.28318530717958647692f / (float)n;
        for (int idx = threadIdx.x; idx < (FFT_N / 2); idx += NT) {
            const int p = idx >> stage;
            const int q = idx & (s - 1);
            float wi, wr;
            sincosf(fac * (float)p, &wi, &wr);           // w = exp(i*sign*2pi*p/n)
            const int   ia = q + s * p;
            const int   ib = ia + (FFT_N / 2);           // s*(n/2) == FFT_N/2 every stage
            const float ar = sR[ia], ai = sI[ia];
            const float br = sR[ib], bi = sI[ib];
            const int   oa = q + 2 * s * p;
            dR[oa]     = ar + br;
            dI[oa]     = ai + bi;
            const float tr = ar - br, ti = ai - bi;
            dR[oa + s] = tr * wr - ti * wi;
            dI[oa + s] = tr * wi + ti * wr;
        }
        __syncthreads();
        n >>= 1;
        float* t;
        t = sR; sR = dR; dR = t;
        t = sI; sI = dI; dI = t;
    }
}

// [A] blocks 0..127: one per (t,c): pad 62x62 -> 64x64, FFT, threshold mask, store k-major.
//     blocks 128..255: zero-fill the padded A rows (t = 8..15) so stage C needs no predication.
extern "C" __global__ __launch_bounds__(NT)
void fftconv_fft_x(const float* __restrict__ x, float* __restrict__ Xr, float* __restrict__ Xi)
{
    __shared__ float sm[4][FFT_N];
    const int row = blockIdx.x;                       // t*16 + c, t in [0,16)
    if (row >= T_DIM * C_DIM) {                       // zero padding rows
        for (int k = threadIdx.x; k < FFT_N; k += NT) {
            Xr[(size_t)k * (T_PAD * C_DIM) + row] = 0.0f;
            Xi[(size_t)k * (T_PAD * C_DIM) + row] = 0.0f;
        }
        return;
    }
    for (int i = threadIdx.x; i < FFT_N; i += NT) {
        const int r = i >> 6, c = i & 63;
        float v = 0.0f;
        if (r >= 1 && r <= OUT_HW && c >= 1 && c <= OUT_HW)
            v = x[((size_t)row * OUT_HW + (r - 1)) * OUT_HW + (c - 1)];
        sm[0][i] = v;
        sm[1][i] = 0.0f;
    }
    block_fft(sm[0], sm[1], sm[2], sm[3], -1.0f);
    for (int k = threadIdx.x; k < FFT_N; k += NT) {
        float re = sm[0][k], im = sm[1][k];
        if (fabsf(re) < 10.0f) { re = 0.0f; im = 0.0f; }
        Xr[(size_t)k * (T_PAD * C_DIM) + row] = re;
        Xi[(size_t)k * (T_PAD * C_DIM) + row] = im;
    }
}

// [B] one block per (o,c): scatter reversed padded W taps, FFT, store k-major [k][o*16+c].
extern "C" __global__ __launch_bounds__(NT)
void fftconv_fft_w(const float* __restrict__ W, float* __restrict__ Wfr, float* __restrict__ Wfi)
{
    __shared__ float sm[4][FFT_N];
    const int o = blockIdx.x >> 4, c = blockIdx.x & 15;
    for (int i = threadIdx.x; i < FFT_N; i += NT) { sm[0][i] = 0.0f; sm[1][i] = 0.0f; }
    __syncthreads();
    if (threadIdx.x < 9) {
        const int kr = threadIdx.x / 3, kc = threadIdx.x % 3;
        // Wp[l] reversed: rev[FFT_N-1-l] = Wp[l], taps at l = kr*64+kc
        sm[0][FFT_N - 1 - (kr * 64 + kc)] = W[(((size_t)o * C_DIM + c) * 3 + kr) * 3 + kc];
    }
    block_fft(sm[0], sm[1], sm[2], sm[3], -1.0f);     // leading barrier covers the scatter
    for (int k = threadIdx.x; k < FFT_N; k += NT) {
        Wfr[(size_t)k * (O_DIM * C_DIM) + o * C_DIM + c] = sm[0][k];
        Wfi[(size_t)k * (O_DIM * C_DIM) + o * C_DIM + c] = sm[1][k];
    }
}

// [C] per-frequency complex contraction: Y[t,o,k] = sum_c X[t,c,k]*W[o,c,k]
// One wave per (k-group of 4, o-half). D = A(16x4 f32, M=t, K=c) x B(4x16 f32, K=c, N=o) + C.
// All operand loads are unconditional aligned b64 (EXEC stays all-ones through the WMMAs).
extern "C" __global__ __launch_bounds__(NT)
void fftconv_contract(const float* __restrict__ Xr, const float* __restrict__ Xi,
                      const float* __restrict__ Wfr, const float* __restrict__ Wfi,
                      float* __restrict__ Yr, float* __restrict__ Yi)
{
    const int gwave = (blockIdx.x * NT + threadIdx.x) >> 5;   // 0 .. 2047
    const int lane  = threadIdx.x & 31;
    const int kbase = (gwave >> 1) * K_PER_WAVE;
    const int o0    = (gwave & 1) << 4;
    const int nn    = lane & 15;     // A: M row; B: N col
    const int hi    = lane >> 4;     // K half select (ISA lane-half K split)

    const int aOff = nn * C_DIM;          // lane offset inside one k-slab of X
    const int bOff = (o0 + nn) * C_DIM;   // lane offset inside one k-slab of Wf

#pragma unroll
    for (int kk = 0; kk < K_PER_WAVE; ++kk) {
        const int k = kbase + kk;
        const float* xr = Xr  + (size_t)k * (T_PAD * C_DIM);
        const float* xi = Xi  + (size_t)k * (T_PAD * C_DIM);
        const float* wr = Wfr + (size_t)k * (O_DIM * C_DIM);
        const float* wi = Wfi + (size_t)k * (O_DIM * C_DIM);
        if (kk + 1 < K_PER_WAVE) {        // warm next frequency slab (global_prefetch_b8)
            __builtin_prefetch(xr + (T_PAD * C_DIM) + aOff, 0, 1);
            __builtin_prefetch(xi + (T_PAD * C_DIM) + aOff, 0, 1);
            __builtin_prefetch(wr + (O_DIM * C_DIM) + bOff, 0, 1);
            __builtin_prefetch(wi + (O_DIM * C_DIM) + bOff, 0, 1);
        }
        v8f accR = {};
        v8f accI = {};
#pragma unroll
        for (int c0 = 0; c0 < C_DIM; c0 += 4) {
            const int kA = c0 + 2 * hi;             // VGPR j holds K = 2*hi + j
            const v2f a_r = ld2(xr + aOff + kA);    // one b64 per operand pair
            const v2f a_i = ld2(xi + aOff + kA);
            const v2f b_r = ld2(wr + bOff + kA);
            const v2f b_i = ld2(wi + bOff + kA);
            v2f a_in; a_in.x = -a_i.x; a_in.y = -a_i.y;  // f32 WMMA has no A-neg modifier
            // Yr += Xr*Wr - Xi*Wi ; Yi += Xr*Wi + Xi*Wr
            accR = __builtin_amdgcn_wmma_f32_16x16x4_f32(false, a_r,  false, b_r, (short)0, accR, false, false);
            accR = __builtin_amdgcn_wmma_f32_16x16x4_f32(false, a_in, false, b_i, (short)0, accR, false, false);
            accI = __builtin_amdgcn_wmma_f32_16x16x4_f32(false, a_r,  false, b_i, (short)0, accI, false, false);
            accI = __builtin_amdgcn_wmma_f32_16x16x4_f32(false, a_i,  false, b_r, (short)0, accI, false, false);
        }
        // D layout: lanes 0-15 hold M = vgpr (t = 0..7); lanes 16-31 hold M = vgpr+8 (padding, drop)
        if (hi == 0) {
            const int o = o0 + nn;
#pragma unroll
            for (int v = 0; v < T_DIM; ++v) {
                Yr[((size_t)(v * O_DIM + o)) * FFT_N + k] = accR[v];
                Yi[((size_t)(v * O_DIM + o)) * FFT_N + k] = accI[v];
            }
        }
    }
}

// [D] one block per (t,o): IFFT, scale 1/N, roll(+1), crop 62x62, add bias.
extern "C" __global__ __launch_bounds__(NT)
void fftconv_ifft_out(const float* __restrict__ Yr, const float* __restrict__ Yi,
                      const float* __restrict__ bias, float* __restrict__ out)
{
    __shared__ float sm[4][FFT_N];
    const int row = blockIdx.x;                 // t*32 + o
    const int o   = row & 31;
    for (int i = threadIdx.x; i < FFT_N; i += NT) {
        sm[0][i] = Yr[(size_t)row * FFT_N + i];
        sm[1][i] = Yi[(size_t)row * FFT_N + i];
    }
    block_fft(sm[0], sm[1], sm[2], sm[3], +1.0f);
    const float bb = bias[o];
    for (int i = threadIdx.x; i < OUT_HW * OUT_HW; i += NT) {
        const int r = i / OUT_HW, c = i - r * OUT_HW;
        const int l = (r * 64 + c + FFT_N - 1) & (FFT_N - 1);   // roll by +1
        out[(size_t)row * (OUT_HW * OUT_HW) + i] = sm[0][l] * (1.0f / FFT_N) + bb;
    }
}

extern "C" void kernel_launch(void* const* d_in, const int* in_sizes, int n_in,
                              void* d_out, int out_size, void* d_ws, size_t ws_size,
                              hipStream_t stream)
{
    const float* x = (const float*)d_in[0];   // (8,16,62,62)
    const float* W = (const float*)d_in[1];   // (32,16,3,3)
    const float* b = (const float*)d_in[2];   // (32,)
    float* out = (float*)d_out;               // (8,32,62,62)

    float* ws  = (float*)d_ws;                // needs 8,388,608 floats = 32 MB
    float* Xr  = ws;
    float* Xi  = Xr  + (size_t)FFT_N * (T_PAD * C_DIM);
    float* Wfr = Xi  + (size_t)FFT_N * (T_PAD * C_DIM);
    float* Wfi = Wfr + (size_t)FFT_N * (O_DIM * C_DIM);
    float* Yr  = Wfi + (size_t)FFT_N * (O_DIM * C_DIM);
    float* Yi  = Yr  + (size_t)(T_DIM * O_DIM) * FFT_N;

    fftconv_fft_x   <<<T_PAD * C_DIM,                      NT, 0, stream>>>(x, Xr, Xi);
    fftconv_fft_w   <<<O_DIM * C_DIM,                      NT, 0, stream>>>(W, Wfr, Wfi);
    fftconv_contract<<<(FFT_N * 2 / K_PER_WAVE) * 32 / NT, NT, 0, stream>>>(Xr, Xi, Wfr, Wfi, Yr, Yi);
    fftconv_ifft_out<<<T_DIM * O_DIM,                      NT, 0, stream>>>(Yr, Yi, b, out);
}